// MoEFFN_18528488915158
// MI455X (gfx1250) — compile-verified
//
#include <hip/hip_runtime.h>
#include <cstdint>

#define D_MODEL   1024
#define D_FF_E    1024
#define D_FFS     2048
#define T_TOK     4096
#define N_EXP     16
#define FFC       256          // ff chunk width for h staging
#define LDX       1028         // padded row stride of X tile (1028 % 64 == 4 -> conflict-free)
#define LDH       260          // padded row stride of h tile

typedef __attribute__((ext_vector_type(2))) float v2f;
typedef __attribute__((ext_vector_type(8))) float v8f;

#if __has_builtin(__builtin_amdgcn_global_load_async_to_lds_b128)
#define HAVE_ASYNC_LDS 1
typedef int v4i_gcc __attribute__((vector_size(16)));
typedef __attribute__((address_space(1))) v4i_gcc* async_gp_t;   // global int4*
typedef __attribute__((address_space(3))) v4i_gcc* async_lp_t;   // LDS int4*
#endif

__device__ __forceinline__ float fast_rcp(float x) {
  return __builtin_amdgcn_rcpf(x);      // v_rcp_f32, ~1 ulp
}

// D(16x16,f32) += A(16x4,f32) * B(4x16,f32), full-precision CDNA5 WMMA
__device__ __forceinline__ v8f wmma_f32_k4(v2f a, v2f b, v8f c) {
  return __builtin_amdgcn_wmma_f32_16x16x4_f32(false, a, false, b, (short)0, c,
                                               false, false);
}

__global__ void zero_counts_kernel(int* cnt) {
  if (threadIdx.x < N_EXP) cnt[threadIdx.x] = 0;
}

// One block per token: 17 dot products (16 router rows + shared gate row),
// top-2 + softmax, scatter token into per-expert lists.
__global__ __launch_bounds__(128)
void router_kernel(const float* __restrict__ X,
                   const float* __restrict__ gw,
                   const float* __restrict__ sgw,
                   int* __restrict__ cnt,
                   int* __restrict__ toks,
                   float* __restrict__ wts,
                   float* __restrict__ sg) {
  __shared__ float red[128];
  __shared__ float logits[17];
  const int t   = blockIdx.x;
  const int tid = threadIdx.x;

  float xv[8];
#pragma unroll
  for (int i = 0; i < 8; i++) xv[i] = X[(size_t)t * D_MODEL + tid + i * 128];

  for (int e = 0; e < 17; e++) {
    const float* wrow = (e < N_EXP) ? (gw + (size_t)e * D_MODEL) : sgw;
    float s = 0.f;
#pragma unroll
    for (int i = 0; i < 8; i++) s += xv[i] * wrow[tid + i * 128];
    red[tid] = s;
    __syncthreads();
    for (int st = 64; st > 0; st >>= 1) {
      if (tid < st) red[tid] += red[tid + st];
      __syncthreads();
    }
    if (tid == 0) logits[e] = red[0];
    __syncthreads();
  }

  if (tid == 0) {
    int i0 = 0; float m0 = logits[0];
    for (int e = 1; e < N_EXP; e++)
      if (logits[e] > m0) { m0 = logits[e]; i0 = e; }
    int i1 = -1; float m1 = -3.4e38f;
    for (int e = 0; e < N_EXP; e++)
      if (e != i0 && logits[e] > m1) { m1 = logits[e]; i1 = e; }
    float e1  = __expf(m1 - m0);
    float inv = fast_rcp(1.f + e1);
    int s0 = atomicAdd(&cnt[i0], 1);
    toks[i0 * T_TOK + s0] = t; wts[i0 * T_TOK + s0] = inv;
    int s1 = atomicAdd(&cnt[i1], 1);
    toks[i1 * T_TOK + s1] = t; wts[i1 * T_TOK + s1] = e1 * inv;
    sg[t] = fast_rcp(1.f + __expf(-logits[16]));
  }
}

// Fused gate_up -> silu*u -> down for one 16-token tile.
// GATHER=true : routed experts (token lists, weight scale, atomicAdd)
// GATHER=false: shared expert (contiguous tokens, sigmoid gate, plain store)
template <int DFFT, bool GATHER>
__global__ __launch_bounds__(256)
void ffn_kernel(const float* __restrict__ X,
                const float* __restrict__ Wgu_all,
                const float* __restrict__ Wd_all,
                const int* __restrict__ cnt,
                const int* __restrict__ toks,
                const float* __restrict__ wts,
                float* __restrict__ out) {
  __shared__ float lds_x[16 * LDX];
  __shared__ float lds_h[16 * LDH];
  __shared__ float lds_w[16];
  __shared__ int   lds_t[16];

  const int tid   = threadIdx.x;
  const int e     = blockIdx.y;
  const int mbase = blockIdx.x * 16;

  const float* Wgu;
  const float* Wd;
  if (GATHER) {
    if (mbase >= cnt[e]) return;     // uniform early-out (before any barrier)
    Wgu = Wgu_all + (size_t)e * (2 * DFFT) * D_MODEL;
    Wd  = Wd_all  + (size_t)e * D_MODEL * DFFT;
  } else {
    Wgu = Wgu_all;
    Wd  = Wd_all;
  }

  if (tid < 16) {
    if (GATHER) {
      const int n_e = cnt[e];
      const int m   = mbase + tid;
      if (m < n_e) { lds_t[tid] = toks[e * T_TOK + m]; lds_w[tid] = wts[e * T_TOK + m]; }
      else         { lds_t[tid] = toks[e * T_TOK + mbase]; lds_w[tid] = 0.f; }
    } else {
      const int t = mbase + tid;
      lds_t[tid] = t;
      lds_w[tid] = wts[t];           // sigmoid(x @ shared_gate_w)
    }
  }
  __syncthreads();

  { // stage X tile: 16 rows x 1024 cols, 16 lanes per row, 16B per transfer
    const int row = tid >> 4;
    const int c16 = tid & 15;
    const float* src = X + (size_t)lds_t[row] * D_MODEL + c16 * 4;
    float* drow = lds_x + row * LDX + c16 * 4;
#ifdef HAVE_ASYNC_LDS
#pragma unroll
    for (int i = 0; i < 16; i++) {
      __builtin_amdgcn_global_load_async_to_lds_b128(
          (async_gp_t)(void*)(const_cast<float*>(src) + 64 * i),
          (async_lp_t)(void*)(drow + 64 * i), 0, 0);
    }
    asm volatile("s_wait_asynccnt 0x0" ::: "memory");
#else
    const float4* src4 = (const float4*)src;
#pragma unroll
    for (int i = 0; i < 16; i++) {
      float4 v = src4[16 * i];
      *(float4*)(drow + 64 * i) = v;
    }
#endif
  }

  const int lane = tid & 31;
  const int wv   = tid >> 5;       // wave id 0..7
  const int kh   = lane >> 4;      // K half-select per A/B f32 fragment layout
  const int l15  = lane & 15;      // row (A) / col (B,D)

  v8f acc[8] = {};                 // 8 x (16x16) output tiles per wave

  for (int ffc = 0; ffc < DFFT; ffc += FFC) {
    __syncthreads();               // also covers X-staging on first iteration
    // ---- phase 1: g,u for this wave's 32 ff columns, silu, h -> LDS ----
#pragma unroll
    for (int p = 0; p < 2; p++) {
      const int ff = ffc + wv * 32 + p * 16;
      const float* Wg   = Wgu + (size_t)(ff + l15) * D_MODEL + 2 * kh;
      const float* Wu   = Wgu + (size_t)(DFFT + ff + l15) * D_MODEL + 2 * kh;
      const float* Arow = lds_x + l15 * LDX + 2 * kh;
      v8f ag = {}, au = {};
#pragma unroll 4
      for (int kb = 0; kb < D_MODEL; kb += 4) {
        v2f A  = *(const v2f*)(Arow + kb);
        v2f Bg = *(const v2f*)(Wg + kb);
        v2f Bu = *(const v2f*)(Wu + kb);
        ag = wmma_f32_k4(A, Bg, ag);
        au = wmma_f32_k4(A, Bu, au);
      }
#pragma unroll
      for (int r = 0; r < 8; r++) {
        float g = ag[r], u = au[r];
        float h = g * fast_rcp(1.f + __expf(-g)) * u;   // silu(g) * u
        lds_h[(r + 8 * kh) * LDH + wv * 32 + p * 16 + l15] = h;
      }
    }
    __syncthreads();
    // ---- phase 2: out[:, wave strip] += h_chunk @ Wd^T ----
#pragma unroll
    for (int j = 0; j < 8; j++) {
      const int nb = wv * 128 + j * 16;
      const float* Wrow = Wd + (size_t)(nb + l15) * DFFT + ffc + 2 * kh;
      const float* Arow = lds_h + l15 * LDH + 2 * kh;
      v8f a = acc[j];
#pragma unroll 4
      for (int kb = 0; kb < FFC; kb += 4) {
        v2f A = *(const v2f*)(Arow + kb);
        v2f B = *(const v2f*)(Wrow + kb);
        a = wmma_f32_k4(A, B, a);
      }
      acc[j] = a;
    }
  }

  // ---- epilogue: D layout VGPR r -> (m = r + 8*kh, n = l15) ----
#pragma unroll
  for (int j = 0; j < 8; j++) {
    const int nb = wv * 128 + j * 16;
#pragma unroll
    for (int r = 0; r < 8; r++) {
      const int m = r + 8 * kh;
      const float val = acc[j][r] * lds_w[m];
      const size_t off = (size_t)lds_t[m] * D_MODEL + nb + l15;
      if (GATHER) unsafeAtomicAdd(out + off, val);
      else        out[off] = val;
    }
  }
}

extern "C" void kernel_launch(void* const* d_in, const int* in_sizes, int n_in,
                              void* d_out, int out_size, void* d_ws, size_t ws_size,
                              hipStream_t stream) {
  (void)in_sizes; (void)n_in; (void)out_size; (void)ws_size;
  const float* X   = (const float*)d_in[0];  // (4096,1024)
  const float* gw  = (const float*)d_in[1];  // (16,1024)
  const float* egu = (const float*)d_in[2];  // (16,2048,1024)
  const float* ed  = (const float*)d_in[3];  // (16,1024,1024)
  const float* sgu = (const float*)d_in[4];  // (4096,1024)
  const float* sd  = (const float*)d_in[5];  // (1024,2048)
  const float* sgw = (const float*)d_in[6];  // (1,1024)
  float* out = (float*)d_out;                // (4096,1024)

  char* ws = (char*)d_ws;
  int*   cnt  = (int*)ws;                                        // 16 ints
  int*   toks = (int*)(ws + 1024);                               // 16*4096 ints
  float* wts  = (float*)(ws + 1024 + N_EXP * T_TOK * 4);         // 16*4096 f32
  float* sg   = (float*)(ws + 1024 + 2 * N_EXP * T_TOK * 4);     // 4096 f32

  zero_counts_kernel<<<1, 32, 0, stream>>>(cnt);
  router_kernel<<<T_TOK, 128, 0, stream>>>(X, gw, sgw, cnt, toks, wts, sg);
  // shared expert first: plain-stores initialize d_out
  ffn_kernel<D_FFS, false><<<dim3(T_TOK / 16, 1), 256, 0, stream>>>(
      X, sgu, sd, nullptr, nullptr, sg, out);
  // routed experts accumulate on top (worst-case grid; idle tiles early-out)
  ffn_kernel<D_FF_E, true><<<dim3(T_TOK / 16, N_EXP), 256, 0, stream>>>(
      X, egu, ed, cnt, toks, wts, out);
}